// DSMIL_48241072668949
// MI455X (gfx1250) — compile-verified
//
#include <hip/hip_runtime.h>
#include <hip/hip_bf16.h>

#define D      1024
#define ADAPT  256
#define NROWS  50000

typedef __attribute__((ext_vector_type(2))) float v2f;
typedef __attribute__((ext_vector_type(8))) float v8f;

extern __shared__ float lds[];

// ---------------- k0: fold Wc into W2:  wc2[k][c] = sum_d W2[k][d]*Wc[d][c]; bc2 = b2@Wc + bc
__global__ __launch_bounds__(256) void k0_fold(const float* __restrict__ W2,
                                               const float* __restrict__ b2,
                                               const float* __restrict__ Wc,
                                               const float* __restrict__ bc,
                                               float* __restrict__ wc2,
                                               float* __restrict__ bc2) {
  const int tid = threadIdx.x;
  for (int idx = tid; idx < ADAPT * 2; idx += 256) {
    int k = idx >> 1, c = idx & 1;
    float s = 0.f;
    for (int d = 0; d < D; ++d) s += W2[(size_t)k * D + d] * Wc[d * 2 + c];
    wc2[idx] = s;
  }
  if (tid < 2) {
    float s = bc[tid];
    for (int d = 0; d < D; ++d) s += b2[d] * Wc[d * 2 + tid];
    bc2[tid] = s;
  }
}

// ---------------- k1: per 16-row tile: h = relu(x@W1+b1); f = h@W2+b2; ilog = h@wc2+bc2
__global__ __launch_bounds__(256) void k1_mlp(const float* __restrict__ x,
                                              const float* __restrict__ W1,
                                              const float* __restrict__ b1,
                                              const float* __restrict__ W2,
                                              const float* __restrict__ b2,
                                              const float* __restrict__ wc2,
                                              const float* __restrict__ bc2,
                                              float* __restrict__ f_out,
                                              float* __restrict__ ilog) {
  float* xs = lds;               // 16 x 1024 (also reused to stage f)
  float* hs = lds + 16 * D;      // 16 x 256
  const int tid  = threadIdx.x;
  const int wave = tid >> 5;
  const int lane = tid & 31;
  const int ml   = lane & 15;
  const int koff = (lane < 16) ? 0 : 2;
  const int mbase = (lane < 16) ? 0 : 8;
  const int row0 = blockIdx.x * 16;

  // async stage x tile -> LDS (ASYNCcnt path; bypasses VGPRs)
  {
    for (int i = tid; i < 16 * D / 4; i += 256) {
      unsigned laddr = (unsigned)(uintptr_t)(xs + (size_t)i * 4);
      const float* gaddr = x + (size_t)row0 * D + (size_t)i * 4;
      asm volatile("global_load_async_to_lds_b128 %0, %1, off"
                   :: "v"(laddr), "v"(gaddr) : "memory");
    }
    asm volatile("s_wait_asynccnt 0x0" ::: "memory");
  }
  __syncthreads();

  // h = relu(x @ W1 + b1): 16 col-tiles, 2 per wave.
  // K-blocked (KB=32): A fragments cached in regs, reused across both tiles;
  // tile-inner loop keeps consecutive wmmas on independent accumulators.
  {
    v8f hacc[2];
    hacc[0] = (v8f){};
    hacc[1] = (v8f){};
    for (int kb = 0; kb < D; kb += 32) {
      v2f afr[8];
#pragma unroll
      for (int kk = 0; kk < 8; ++kk) {
        afr[kk].x = xs[ml * D + kb + kk * 4 + koff];
        afr[kk].y = xs[ml * D + kb + kk * 4 + koff + 1];
      }
#pragma unroll
      for (int kk = 0; kk < 8; ++kk) {
#pragma unroll
        for (int t = 0; t < 2; ++t) {
          const int n0 = (wave * 2 + t) * 16;
          v2f b;
          b.x = W1[(size_t)(kb + kk * 4 + koff)     * ADAPT + n0 + ml];
          b.y = W1[(size_t)(kb + kk * 4 + koff + 1) * ADAPT + n0 + ml];
          hacc[t] = __builtin_amdgcn_wmma_f32_16x16x4_f32(false, afr[kk], false, b,
                                                          (short)0, hacc[t], false, false);
        }
      }
    }
#pragma unroll
    for (int t = 0; t < 2; ++t) {
      const int n = (wave * 2 + t) * 16 + ml;
      const float bias = b1[n];
#pragma unroll
      for (int j = 0; j < 8; ++j) {
        float v = hacc[t][j] + bias;
        hs[(j + mbase) * ADAPT + n] = v > 0.f ? v : 0.f;
      }
    }
  }
  __syncthreads();

  // f = h @ W2 + b2: 64 col-tiles, 8 per wave; K-blocked, staged into xs.
  {
    v8f acc[8];
#pragma unroll
    for (int t = 0; t < 8; ++t) acc[t] = (v8f){};
    for (int kb = 0; kb < ADAPT; kb += 32) {
      v2f afr[8];
#pragma unroll
      for (int kk = 0; kk < 8; ++kk) {
        afr[kk].x = hs[ml * ADAPT + kb + kk * 4 + koff];
        afr[kk].y = hs[ml * ADAPT + kb + kk * 4 + koff + 1];
      }
#pragma unroll
      for (int kk = 0; kk < 8; ++kk) {
#pragma unroll
        for (int t = 0; t < 8; ++t) {
          const int n0 = wave * 128 + t * 16;
          v2f b;
          b.x = W2[(size_t)(kb + kk * 4 + koff)     * D + n0 + ml];
          b.y = W2[(size_t)(kb + kk * 4 + koff + 1) * D + n0 + ml];
          acc[t] = __builtin_amdgcn_wmma_f32_16x16x4_f32(false, afr[kk], false, b,
                                                         (short)0, acc[t], false, false);
        }
      }
    }
#pragma unroll
    for (int t = 0; t < 8; ++t) {
      const int n = wave * 128 + t * 16 + ml;
      const float bias = b2[n];
#pragma unroll
      for (int j = 0; j < 8; ++j) xs[(j + mbase) * D + n] = acc[t][j] + bias;
    }
  }

  // instance logits from hidden layer (wave 0; only cols 0..1 valid/stored)
  if (wave == 0) {
    v8f acc = {};
    const int nc = (ml < 2) ? ml : 0;
    for (int k = 0; k < ADAPT; k += 4) {
      v2f a, b;
      a.x = hs[ml * ADAPT + k + koff];
      a.y = hs[ml * ADAPT + k + koff + 1];
      b.x = wc2[(k + koff) * 2 + nc];
      b.y = wc2[(k + koff + 1) * 2 + nc];
      acc = __builtin_amdgcn_wmma_f32_16x16x4_f32(false, a, false, b, (short)0, acc, false, false);
    }
    if (ml < 2) {
      const float bias = bc2[ml];
#pragma unroll
      for (int j = 0; j < 8; ++j)
        ilog[(size_t)(row0 + j + mbase) * 2 + ml] = acc[j] + bias;
    }
  }
  __syncthreads();

  // async f writeback LDS -> global (ASYNCcnt path)
  {
    for (int i = tid; i < 16 * D / 4; i += 256) {
      unsigned laddr = (unsigned)(uintptr_t)(xs + (size_t)i * 4);
      float* gaddr = f_out + (size_t)row0 * D + (size_t)i * 4;
      asm volatile("global_store_async_from_lds_b128 %0, %1, off"
                   :: "v"(gaddr), "v"(laddr) : "memory");
    }
    asm volatile("s_wait_asynccnt 0x0" ::: "memory");
  }
}

// ---------------- k2: argmax + max of instance logits over N (per class)
__global__ __launch_bounds__(256) void k2_argmax(const float* __restrict__ ilog,
                                                 int* __restrict__ topidx,
                                                 float* __restrict__ maxil) {
  __shared__ float sv[2][256];
  __shared__ int   si[2][256];
  const int tid = threadIdx.x;
  float b0 = -3.4e38f, b1 = -3.4e38f;
  int i0 = 0, i1 = 0;
  for (int i = tid; i < NROWS; i += 256) {
    float v0 = ilog[(size_t)i * 2], v1 = ilog[(size_t)i * 2 + 1];
    if (v0 > b0) { b0 = v0; i0 = i; }
    if (v1 > b1) { b1 = v1; i1 = i; }
  }
  sv[0][tid] = b0; si[0][tid] = i0;
  sv[1][tid] = b1; si[1][tid] = i1;
  __syncthreads();
  for (int off = 128; off > 0; off >>= 1) {
    if (tid < off) {
      for (int c = 0; c < 2; ++c) {
        float va = sv[c][tid], vb = sv[c][tid + off];
        int ia = si[c][tid], ib = si[c][tid + off];
        if (vb > va || (vb == va && ib < ia)) { sv[c][tid] = vb; si[c][tid] = ib; }
      }
    }
    __syncthreads();
  }
  if (tid == 0) {
    topidx[0] = si[0][0]; topidx[1] = si[1][0];
    maxil[0] = sv[0][0];  maxil[1] = sv[1][0];
  }
}

// ---------------- k3: r_c = Wq @ (f[top_c]@Wq + bq),  beta_c = bq . q_top_c
__global__ __launch_bounds__(256) void k3_rvec(const float* __restrict__ f,
                                               const float* __restrict__ Wq,
                                               const float* __restrict__ bq,
                                               const int* __restrict__ topidx,
                                               float* __restrict__ rvec,
                                               float* __restrict__ beta) {
  __shared__ float ftop[2][D];
  __shared__ float tql[2][384];
  const int tid = threadIdx.x;
  const int t0 = topidx[0], t1 = topidx[1];
  for (int d = tid; d < D; d += 256) {
    ftop[0][d] = f[(size_t)t0 * D + d];
    ftop[1][d] = f[(size_t)t1 * D + d];
  }
  __syncthreads();
  for (int idx = tid; idx < 2 * 384; idx += 256) {
    int c = idx / 384, a = idx % 384;
    float s = bq[a];
    for (int d = 0; d < D; ++d) s += ftop[c][d] * Wq[(size_t)d * 384 + a];
    tql[c][a] = s;
  }
  __syncthreads();
  for (int idx = tid; idx < 2 * D; idx += 256) {
    int c = idx >> 10, d = idx & (D - 1);
    float s = 0.f;
    for (int a = 0; a < 384; ++a) s += Wq[(size_t)d * 384 + a] * tql[c][a];
    rvec[idx] = s;
  }
  if (tid < 2) {
    float s = 0.f;
    for (int a = 0; a < 384; ++a) s += bq[a] * tql[tid][a];
    beta[tid] = s;
  }
}

// ---------------- k4: scores[i][c] = f_i . r_c + beta_c ; score_vec = scores[:,0]
__global__ __launch_bounds__(256) void k4_scores(const float* __restrict__ f,
                                                 const float* __restrict__ rvec,
                                                 const float* __restrict__ beta,
                                                 float* __restrict__ scores,
                                                 float* __restrict__ score_out) {
  __shared__ float rl[2 * D];
  __shared__ float bl[2];
  const int tid = threadIdx.x;
  const int wave = tid >> 5;
  const int lane = tid & 31;
  for (int i = tid; i < 2 * D; i += 256) rl[i] = rvec[i];
  if (tid < 2) bl[tid] = beta[tid];
  __syncthreads();
  const int base = blockIdx.x * 64;
  for (int rr = 0; rr < 8; ++rr) {
    const int row = base + wave * 8 + rr;
    if (row >= NROWS) continue;
    const float* frow = f + (size_t)row * D;
    float s0 = 0.f, s1 = 0.f;
    for (int d = lane; d < D; d += 32) {
      float fv = frow[d];
      s0 += fv * rl[d];
      s1 += fv * rl[D + d];
    }
    for (int off = 16; off > 0; off >>= 1) {
      s0 += __shfl_down(s0, off, 32);
      s1 += __shfl_down(s1, off, 32);
    }
    if (lane == 0) {
      s0 += bl[0]; s1 += bl[1];
      scores[(size_t)row * 2]     = s0;
      scores[(size_t)row * 2 + 1] = s1;
      score_out[row] = s0;   // raw attention_scores[:,0]
    }
  }
}

// ---------------- k5: softmax stats over N (per class) + zero g
__global__ __launch_bounds__(256) void k5_stats(const float* __restrict__ scores,
                                                float* __restrict__ stats,
                                                float* __restrict__ g) {
  __shared__ float r0[256], r1[256];
  __shared__ float mm[2];
  const int tid = threadIdx.x;
  const float isq = 0.05103103630798288f;  // 1/sqrt(384)
  float m0 = -3.4e38f, m1 = -3.4e38f;
  for (int i = tid; i < NROWS; i += 256) {
    m0 = fmaxf(m0, scores[(size_t)i * 2] * isq);
    m1 = fmaxf(m1, scores[(size_t)i * 2 + 1] * isq);
  }
  r0[tid] = m0; r1[tid] = m1;
  __syncthreads();
  for (int off = 128; off > 0; off >>= 1) {
    if (tid < off) { r0[tid] = fmaxf(r0[tid], r0[tid + off]); r1[tid] = fmaxf(r1[tid], r1[tid + off]); }
    __syncthreads();
  }
  if (tid == 0) { mm[0] = r0[0]; mm[1] = r1[0]; stats[0] = r0[0]; stats[1] = r1[0]; }
  __syncthreads();
  const float M0 = mm[0], M1 = mm[1];
  float s0 = 0.f, s1 = 0.f;
  for (int i = tid; i < NROWS; i += 256) {
    s0 += expf(scores[(size_t)i * 2] * isq - M0);
    s1 += expf(scores[(size_t)i * 2 + 1] * isq - M1);
  }
  r0[tid] = s0; r1[tid] = s1;
  __syncthreads();
  for (int off = 128; off > 0; off >>= 1) {
    if (tid < off) { r0[tid] += r0[tid + off]; r1[tid] += r1[tid + off]; }
    __syncthreads();
  }
  if (tid == 0) { stats[2] = r0[0]; stats[3] = r1[0]; }
  for (int i = tid; i < 2 * D; i += 256) g[i] = 0.f;
}

// ---------------- k6: g_c = sum_i w_ic * f_i  (chunked, f32 global atomics)
__global__ __launch_bounds__(256) void k6_agg(const float* __restrict__ f,
                                              const float* __restrict__ scores,
                                              const float* __restrict__ stats,
                                              float* __restrict__ g) {
  __shared__ float wl[256][2];
  __shared__ float sm[4];
  const int tid = threadIdx.x;
  const float isq = 0.05103103630798288f;
  if (tid < 4) sm[tid] = stats[tid];
  __syncthreads();
  const float M0 = sm[0], M1 = sm[1];
  const float iZ0 = 1.f / sm[2], iZ1 = 1.f / sm[3];
  const int base = blockIdx.x * 256;
  const int row = base + tid;
  float w0 = 0.f, w1 = 0.f;
  if (row < NROWS) {
    w0 = expf(scores[(size_t)row * 2] * isq - M0) * iZ0;
    w1 = expf(scores[(size_t)row * 2 + 1] * isq - M1) * iZ1;
  }
  wl[tid][0] = w0; wl[tid][1] = w1;
  __syncthreads();
  const int nrows = (NROWS - base) < 256 ? (NROWS - base) : 256;
  float acc0[4] = {}, acc1[4] = {};
  for (int i = 0; i < nrows; ++i) {
    const float* frow = f + (size_t)(base + i) * D;
    const float a0 = wl[i][0], a1 = wl[i][1];
#pragma unroll
    for (int jj = 0; jj < 4; ++jj) {
      float fv = frow[tid + 256 * jj];
      acc0[jj] += a0 * fv;
      acc1[jj] += a1 * fv;
    }
  }
#pragma unroll
  for (int jj = 0; jj < 4; ++jj) {
    atomicAdd(&g[tid + 256 * jj], acc0[jj]);
    atomicAdd(&g[D + tid + 256 * jj], acc1[jj]);
  }
}

// ---------------- k7: bag = g@Wv + bv; layernorm; conv einsum; final logits
__global__ __launch_bounds__(256) void k7_final(const float* __restrict__ g,
                                                const float* __restrict__ Wv,
                                                const float* __restrict__ bv,
                                                const float* __restrict__ ln_g,
                                                const float* __restrict__ ln_b,
                                                const float* __restrict__ conv_w,
                                                const float* __restrict__ conv_b,
                                                const float* __restrict__ maxil,
                                                float* __restrict__ out) {
  __shared__ float bag[2][D];
  __shared__ float red[256];
  __shared__ float sc[2];
  const int tid = threadIdx.x;
  for (int idx = tid; idx < 2 * D; idx += 256) {
    int c = idx >> 10, d = idx & (D - 1);
    float s = bv[d];
    const float* gc = g + c * D;
    for (int k = 0; k < D; ++k) s += gc[k] * Wv[(size_t)k * D + d];
    bag[c][d] = s;
  }
  __syncthreads();
  for (int c = 0; c < 2; ++c) {
    float p = 0.f;
    for (int d = tid; d < D; d += 256) p += bag[c][d];
    red[tid] = p; __syncthreads();
    for (int off = 128; off > 0; off >>= 1) { if (tid < off) red[tid] += red[tid + off]; __syncthreads(); }
    if (tid == 0) sc[0] = red[0] * (1.0f / D);
    __syncthreads();
    const float mu = sc[0];
    p = 0.f;
    for (int d = tid; d < D; d += 256) { float t = bag[c][d] - mu; p += t * t; }
    red[tid] = p; __syncthreads();
    for (int off = 128; off > 0; off >>= 1) { if (tid < off) red[tid] += red[tid + off]; __syncthreads(); }
    if (tid == 0) sc[1] = rsqrtf(red[0] * (1.0f / D) + 1e-5f);
    __syncthreads();
    const float rs = sc[1];
    for (int d = tid; d < D; d += 256) bag[c][d] = (bag[c][d] - mu) * rs * ln_g[d] + ln_b[d];
    __syncthreads();
  }
  for (int o = 0; o < 2; ++o) {
    float p = 0.f;
    for (int idx = tid; idx < 2 * D; idx += 256)
      p += bag[idx >> 10][idx & (D - 1)] * conv_w[(size_t)o * 2 * D + idx];
    red[tid] = p; __syncthreads();
    for (int off = 128; off > 0; off >>= 1) { if (tid < off) red[tid] += red[tid + off]; __syncthreads(); }
    if (tid == 0) out[o] = 0.5f * ((red[0] + conv_b[o]) + maxil[o]);
    __syncthreads();
  }
}

extern "C" void kernel_launch(void* const* d_in, const int* in_sizes, int n_in,
                              void* d_out, int out_size, void* d_ws, size_t ws_size,
                              hipStream_t stream) {
  const float* x    = (const float*)d_in[0];
  const float* W1   = (const float*)d_in[1];
  const float* b1   = (const float*)d_in[2];
  const float* W2   = (const float*)d_in[3];
  const float* b2   = (const float*)d_in[4];
  const float* Wc   = (const float*)d_in[5];
  const float* bc   = (const float*)d_in[6];
  const float* Wq   = (const float*)d_in[7];
  const float* bq   = (const float*)d_in[8];
  const float* Wv   = (const float*)d_in[9];
  const float* bv   = (const float*)d_in[10];
  const float* ln_g = (const float*)d_in[11];
  const float* ln_b = (const float*)d_in[12];
  const float* cw   = (const float*)d_in[13];
  const float* cb   = (const float*)d_in[14];
  float* out = (float*)d_out;

  float* ws = (float*)d_ws;
  size_t off = 0;
  float* f      = ws + off; off += (size_t)NROWS * D;
  float* ilog   = ws + off; off += (size_t)NROWS * 2;
  float* scores = ws + off; off += (size_t)NROWS * 2;
  float* wc2    = ws + off; off += ADAPT * 2;
  float* bc2    = ws + off; off += 2;
  float* maxil  = ws + off; off += 2;
  float* stats  = ws + off; off += 4;
  float* rvec   = ws + off; off += 2 * D;
  float* beta   = ws + off; off += 2;
  float* g      = ws + off; off += 2 * D;
  int* topidx   = (int*)(ws + off); off += 2;

  const size_t lds1 = (size_t)(16 * D + 16 * ADAPT) * sizeof(float);  // 80 KB

  k0_fold<<<1, 256, 0, stream>>>(W2, b2, Wc, bc, wc2, bc2);
  k1_mlp<<<NROWS / 16, 256, lds1, stream>>>(x, W1, b1, W2, b2, wc2, bc2, f, ilog);
  k2_argmax<<<1, 256, 0, stream>>>(ilog, topidx, maxil);
  k3_rvec<<<1, 256, 0, stream>>>(f, Wq, bq, topidx, rvec, beta);
  k4_scores<<<(NROWS + 63) / 64, 256, 0, stream>>>(f, rvec, beta, scores, out + 2);
  k5_stats<<<1, 256, 0, stream>>>(scores, stats, g);
  k6_agg<<<(NROWS + 255) / 256, 256, 0, stream>>>(f, scores, stats, g);
  k7_final<<<1, 256, 0, stream>>>(g, Wv, bv, ln_g, ln_b, cw, cb, maxil, out);
}